// HashGridLoRAEncoder_70420283785503
// MI455X (gfx1250) — compile-verified
//
#include <hip/hip_runtime.h>
#include <stdint.h>

#define NLEVELS 16
#define HASHSZ  32768
#define NFEAT   2
#define TABLE_FLOATS (HASHSZ * NFEAT)   /* 65536 floats = 256 KB per level */
#define BLOCK   1024
#define CHUNK_PTS 16384                  /* points per workgroup */

__constant__ int kRes[NLEVELS] = {16, 20, 25, 32, 40, 50, 64, 80,
                                  101, 128, 161, 203, 256, 322, 406, 512};

__global__ __launch_bounds__(BLOCK)
void hashgrid_encode_kernel(const float* __restrict__ x,
                            const float* __restrict__ tables,
                            float* __restrict__ out, int n, int chunk_pts) {
    __shared__ __align__(16) float tab[TABLE_FLOATS];

    const int tid   = threadIdx.x;
    const int lvl   = blockIdx.x & (NLEVELS - 1);   // interleave levels across WGs
    const int chunk = blockIdx.x >> 4;

    // ---- Stage this level's full 256 KB table into LDS (once per WG) -------
    {
        const float*   src      = tables + (size_t)lvl * TABLE_FLOATS;
        const uint32_t lds_base = (uint32_t)(uintptr_t)&tab[0];
#pragma unroll 1
        for (int i = tid; i < TABLE_FLOATS / 4; i += BLOCK) {
            uint64_t ga = (uint64_t)(uintptr_t)(src + (size_t)i * 4);
            uint32_t la = lds_base + (uint32_t)i * 16u;
            asm volatile("global_load_async_to_lds_b128 %0, %1, off"
                         :: "v"(la), "v"(ga) : "memory");
        }
#if __has_builtin(__builtin_amdgcn_s_wait_asynccnt)
        __builtin_amdgcn_s_wait_asynccnt(0);
#else
        asm volatile("s_wait_asynccnt 0x0" ::: "memory");
#endif
        __syncthreads();
    }

    // ---- Stream this WG's point chunk against the staged table -------------
    const float   res  = (float)kRes[lvl];
    const float2* t2   = (const float2*)tab;
    float2*       out2 = (float2*)out;

    int p0   = chunk * chunk_pts;
    int pend = p0 + chunk_pts;
    if (pend > n) pend = n;

#pragma unroll 1
    for (int p = p0 + tid; p < pend; p += BLOCK) {
        const float* xp = x + (size_t)p * 3;
        // Match reference op order exactly: ((x+1)*0.5) * res, floor, frac.
        float xs = ((xp[0] + 1.0f) * 0.5f) * res;
        float ys = ((xp[1] + 1.0f) * 0.5f) * res;
        float zs = ((xp[2] + 1.0f) * 0.5f) * res;
        float flx = floorf(xs), fly = floorf(ys), flz = floorf(zs);
        float fx = xs - flx, fy = ys - fly, fz = zs - flz;
        uint32_t xi = (uint32_t)flx;
        uint32_t yi = (uint32_t)fly;
        uint32_t zi = (uint32_t)flz;

        uint32_t hx0 = xi;                        // prime 1
        uint32_t hx1 = xi + 1u;
        uint32_t hy0 = yi * 2654435761u;
        uint32_t hy1 = hy0 + 2654435761u;         // (yi+1)*p1 with uint32 wrap
        uint32_t hz0 = zi * 805459861u;
        uint32_t hz1 = hz0 + 805459861u;

        float wx0 = 1.0f - fx, wy0 = 1.0f - fy, wz0 = 1.0f - fz;

        float f0 = 0.0f, f1 = 0.0f;
#pragma unroll
        for (int c = 0; c < 8; ++c) {
            uint32_t h = ((c & 1) ? hx1 : hx0) ^
                         ((c & 2) ? hy1 : hy0) ^
                         ((c & 4) ? hz1 : hz0);
            uint32_t idx = h & (uint32_t)(HASHSZ - 1);
            float w = ((c & 1) ? fx : wx0) *
                      ((c & 2) ? fy : wy0) *
                      ((c & 4) ? fz : wz0);
            float2 f = t2[idx];                    // ds_load_b64 gather
            f0 += w * f.x;
            f1 += w * f.y;
        }
        // One point's row (32 floats) == one 128B line; 16 concurrent level-WGs
        // assemble it in L2, single writeback to HBM.
        out2[(size_t)p * NLEVELS + lvl] = make_float2(f0, f1);
    }
}

extern "C" void kernel_launch(void* const* d_in, const int* in_sizes, int n_in,
                              void* d_out, int out_size, void* d_ws, size_t ws_size,
                              hipStream_t stream) {
    const float* x      = (const float*)d_in[0];   // (N, 3) float32
    const float* tables = (const float*)d_in[1];   // (16, 32768, 2) float32
    float*       out    = (float*)d_out;           // (N, 32) float32
    (void)d_ws; (void)ws_size; (void)n_in; (void)out_size;

    int n = in_sizes[0] / 3;
    if (n <= 0) return;
    int nchunks = (n + CHUNK_PTS - 1) / CHUNK_PTS;
    int grid    = nchunks * NLEVELS;
    hashgrid_encode_kernel<<<grid, BLOCK, 0, stream>>>(x, tables, out, n, CHUNK_PTS);
}